// Model_41266045780522
// MI455X (gfx1250) — compile-verified
//
#include <hip/hip_runtime.h>
#include <hip/hip_bf16.h>

typedef __attribute__((ext_vector_type(2))) float v2f;
typedef __attribute__((ext_vector_type(8))) float v8f;

#define NB   16
#define CIN  3
#define HH   512
#define WWI  512
#define PD   27
#define HID  64
#define COUT 4
#define DD   36
#define LL   65536

// ---- workspace layout (floats) ----
// [0,432)      inst 'a'  (rstd*inst_w)  per (n,pd)
// [432,864)    inst 'b'  (inst_b - mean*a)
// [864,928)    bn_sum[64]
// [928,992)    bn_sumsq[64]
// [992,1056)   bn_scale[64]
// [1056,1120)  bn_bias[64]
// [1120,1136)  softmax max per n
// [1136,1152)  softmax sumexp per n
// [1152, 1152+16*65536) scores

// patch value with folded instance-norm affine: pn = x*a + b
// s_geom[pd] = c | ki<<8 | kj<<16 | invalid<<24  (pd==27 slot is invalid, a=b=0)
__device__ __forceinline__ float patch_val(const float* __restrict__ x,
                                           const float* __restrict__ s_wa,
                                           const float* __restrict__ s_wb,
                                           const int* __restrict__ s_geom,
                                           int n, int pd, int ho, int wo) {
  int g  = s_geom[pd];
  int c  = g & 0xff, ki = (g >> 8) & 0xff, kj = (g >> 16) & 0xff;
  int hh = ki + 2 * ho - 1, ww = kj + 2 * wo - 1;
  float xv = 0.0f;
  bool inb = (hh >= 0) && (hh < HH) && (ww >= 0) && (ww < WWI) && ((g >> 24) == 0);
  if (inb) xv = x[((n * CIN + c) * HH + hh) * WWI + ww];
  return fmaf(xv, s_wa[pd], s_wb[pd]);
}

// Fill LDS tables shared by the WMMA kernels. Must be followed by __syncthreads().
__device__ __forceinline__ void stage_common(const float* __restrict__ proj_w,
                                             const float* __restrict__ ws,
                                             int n,
                                             v2f* __restrict__ sAfrag,   // [4*7*32]
                                             float* __restrict__ s_wa,   // [28]
                                             float* __restrict__ s_wb,   // [28]
                                             int* __restrict__ s_geom) { // [28]
  for (int i = threadIdx.x; i < 28; i += blockDim.x) {
    if (i < PD) {
      s_wa[i] = ws[n * PD + i];
      s_wb[i] = ws[432 + n * PD + i];
      int c = i / 9, kk = i % 9;
      s_geom[i] = c | ((kk / 3) << 8) | ((kk % 3) << 16);
    } else {
      s_wa[i] = 0.0f; s_wb[i] = 0.0f; s_geom[i] = 1 << 24;
    }
  }
  // A fragments: lane ln holds row h = t*16 + (ln&15), k-pair k0 = 4j + 2*(ln>>4)
  for (int idx = threadIdx.x; idx < 4 * 7 * 32; idx += blockDim.x) {
    int t = idx / 224, rem = idx % 224, j = rem / 32, ln = rem % 32;
    int h  = t * 16 + (ln & 15);
    int k0 = 4 * j + 2 * (ln >> 4);
    v2f v;
    v.x = (k0     < PD) ? proj_w[h * PD + k0]     : 0.0f;
    v.y = (k0 + 1 < PD) ? proj_w[h * PD + k0 + 1] : 0.0f;
    sAfrag[idx] = v;
  }
}

// One 64(h) x 16(l) hidden tile via f32 WMMA; acc[t] = D tile for h in [16t,16t+16).
__device__ __forceinline__ void hidden_tile(const float* __restrict__ x,
                                            const v2f* __restrict__ sAfrag,
                                            const float* __restrict__ s_wa,
                                            const float* __restrict__ s_wb,
                                            const int* __restrict__ s_geom,
                                            int n, int lbase, int lane, int hi,
                                            v8f (&acc)[4]) {
  int l = lbase + (lane & 15);
  int ho = l >> 8, wo = l & 255;
  v8f z = 0.0f;
  acc[0] = z; acc[1] = z; acc[2] = z; acc[3] = z;
#pragma unroll
  for (int j = 0; j < 7; ++j) {
    int pd0 = 4 * j + 2 * hi;                 // B operand: lane low16 = column l
    float b0 = patch_val(x, s_wa, s_wb, s_geom, n, pd0,     ho, wo);
    float b1 = patch_val(x, s_wa, s_wb, s_geom, n, pd0 + 1, ho, wo);
    v2f bv = {b0, b1};
#pragma unroll
    for (int t = 0; t < 4; ++t) {
      v2f av = sAfrag[(t * 7 + j) * 32 + lane];   // ds_load_b64
      acc[t] = __builtin_amdgcn_wmma_f32_16x16x4_f32(
          false, av, false, bv, (short)0, acc[t], false, false);
    }
  }
}

// ---------------- pass 0: zero output + BN accumulators ----------------
__global__ void k0_zero(float* __restrict__ out, int nout, float* __restrict__ ws) {
  for (int i = blockIdx.x * blockDim.x + threadIdx.x; i < nout;
       i += gridDim.x * blockDim.x)
    out[i] = 0.0f;
  if (blockIdx.x == 0 && threadIdx.x < 128) ws[864 + threadIdx.x] = 0.0f;
}

// ---------------- pass 1: instance-norm stats -> folded a,b ----------------
__global__ void k1_inst(const float* __restrict__ x, const float* __restrict__ inst_w,
                        const float* __restrict__ inst_b, float* __restrict__ ws) {
  __shared__ float r0[256], r1[256];
  int id = blockIdx.x;                        // n*27 + pd
  int n = id / PD, pd = id % PD;
  int c = pd / 9, kk = pd % 9, ki = kk / 3, kj = kk % 3;
  float su = 0.0f, sq = 0.0f;
  for (int l = threadIdx.x; l < LL; l += 256) {
    int ho = l >> 8, wo = l & 255;
    int hh = ki + 2 * ho - 1, ww = kj + 2 * wo - 1;
    float v = 0.0f;
    if (hh >= 0 && hh < HH && ww >= 0 && ww < WWI)
      v = x[((n * CIN + c) * HH + hh) * WWI + ww];
    su += v; sq += v * v;
  }
  r0[threadIdx.x] = su; r1[threadIdx.x] = sq;
  __syncthreads();
  for (int off = 128; off > 0; off >>= 1) {
    if (threadIdx.x < off) {
      r0[threadIdx.x] += r0[threadIdx.x + off];
      r1[threadIdx.x] += r1[threadIdx.x + off];
    }
    __syncthreads();
  }
  if (threadIdx.x == 0) {
    float mean = r0[0] * (1.0f / LL);
    float var  = r1[0] * (1.0f / LL) - mean * mean;
    float a = rsqrtf(var + 1e-5f) * inst_w[pd];
    ws[n * PD + pd]       = a;
    ws[432 + n * PD + pd] = inst_b[pd] - mean * a;
  }
}

// ---------------- pass 2: hidden (WMMA) -> BN sum/sumsq ----------------
__global__ __launch_bounds__(256, 1)
void k2_bnstats(const float* __restrict__ x, const float* __restrict__ proj_w,
                float* __restrict__ ws) {
  __shared__ v2f   sAfrag[4 * 7 * 32];
  __shared__ float s_wa[28], s_wb[28];
  __shared__ int   s_geom[28];
  int lane = threadIdx.x & 31, wave = threadIdx.x >> 5;
  int hi = lane >> 4;
  int n = blockIdx.x >> 6, chunk = blockIdx.x & 63;   // 64 blocks per n
  stage_common(proj_w, ws, n, sAfrag, s_wa, s_wb, s_geom);
  __syncthreads();
  float* bn_sum = ws + 864;
  float* bn_sq  = ws + 928;
  float sA[4][8], sQ[4][8];
#pragma unroll
  for (int t = 0; t < 4; ++t)
#pragma unroll
    for (int r = 0; r < 8; ++r) { sA[t][r] = 0.0f; sQ[t][r] = 0.0f; }
  for (int it = 0; it < 8; ++it) {
    int lbase = ((chunk * 8 + wave) * 8 + it) * 16;
    v8f acc[4];
    hidden_tile(x, sAfrag, s_wa, s_wb, s_geom, n, lbase, lane, hi, acc);
#pragma unroll
    for (int t = 0; t < 4; ++t)
#pragma unroll
      for (int r = 0; r < 8; ++r) {
        float v = acc[t][r];
        sA[t][r] += v; sQ[t][r] += v * v;
      }
  }
#pragma unroll
  for (int t = 0; t < 4; ++t)
#pragma unroll
    for (int r = 0; r < 8; ++r)
      for (int off = 1; off < 16; off <<= 1) {
        sA[t][r] += __shfl_xor(sA[t][r], off, 32);
        sQ[t][r] += __shfl_xor(sQ[t][r], off, 32);
      }
  if ((lane & 15) == 0) {
#pragma unroll
    for (int t = 0; t < 4; ++t)
#pragma unroll
      for (int r = 0; r < 8; ++r) {
        int h = t * 16 + 8 * hi + r;
        atomicAdd(&bn_sum[h], sA[t][r]);
        atomicAdd(&bn_sq[h],  sQ[t][r]);
      }
  }
}

// ---------------- pass 2b: finalize BN scale/bias ----------------
__global__ void k2b_bnfin(const float* __restrict__ bn_w, const float* __restrict__ bn_b,
                          float* __restrict__ ws) {
  int h = threadIdx.x;
  if (h < HID) {
    float cnt = (float)NB * (float)LL;
    float mean = ws[864 + h] / cnt;
    float var  = ws[928 + h] / cnt - mean * mean;
    float sc = bn_w[h] * rsqrtf(var + 1e-5f);
    ws[992 + h]  = sc;
    ws[1056 + h] = bn_b[h] - mean * sc;
  }
}

// ---------------- pass 3: hidden (WMMA) -> attention scores ----------------
__global__ __launch_bounds__(256, 1)
void k3_scores(const float* __restrict__ x, const float* __restrict__ proj_w,
               const float* __restrict__ scorer_w, const float* __restrict__ scorer_b,
               float* __restrict__ ws) {
  __shared__ v2f   sAfrag[4 * 7 * 32];
  __shared__ float s_wa[28], s_wb[28];
  __shared__ int   s_geom[28];
  __shared__ float sscale[HID], sbias[HID], ssw[HID];
  int lane = threadIdx.x & 31, wave = threadIdx.x >> 5;
  int hi = lane >> 4;
  int n = blockIdx.x >> 6, chunk = blockIdx.x & 63;
  stage_common(proj_w, ws, n, sAfrag, s_wa, s_wb, s_geom);
  for (int i = threadIdx.x; i < HID; i += blockDim.x) {
    sscale[i] = ws[992 + i];
    sbias[i]  = ws[1056 + i];
    ssw[i]    = scorer_w[i];
  }
  __syncthreads();
  float* scores = ws + 1152;
  float sb = scorer_b[0];
  for (int it = 0; it < 8; ++it) {
    int lbase = ((chunk * 8 + wave) * 8 + it) * 16;
    v8f acc[4];
    hidden_tile(x, sAfrag, s_wa, s_wb, s_geom, n, lbase, lane, hi, acc);
    float sp = 0.0f;
#pragma unroll
    for (int t = 0; t < 4; ++t)
#pragma unroll
      for (int r = 0; r < 8; ++r) {
        int h = t * 16 + 8 * hi + r;
        float v = fmaf(acc[t][r], sscale[h], sbias[h]);
        v = fmaxf(v, 0.0f);
        sp = fmaf(v, ssw[h], sp);
      }
    float tot = sp + __shfl_xor(sp, 16, 32);   // combine h halves of column l
    if (hi == 0) scores[n * LL + lbase + (lane & 15)] = tot + sb;
  }
}

// ---------------- pass 4: softmax stats per n ----------------
__global__ void k4_softmax(float* __restrict__ ws) {
  __shared__ float red[256];
  int n = blockIdx.x;
  const float* sc = ws + 1152 + n * LL;
  float m = -1e30f;
  for (int i = threadIdx.x; i < LL; i += 256) m = fmaxf(m, sc[i]);
  red[threadIdx.x] = m;
  __syncthreads();
  for (int off = 128; off > 0; off >>= 1) {
    if (threadIdx.x < off)
      red[threadIdx.x] = fmaxf(red[threadIdx.x], red[threadIdx.x + off]);
    __syncthreads();
  }
  float mx = red[0];
  __syncthreads();
  float su = 0.0f;
  for (int i = threadIdx.x; i < LL; i += 256) su += __expf(sc[i] - mx);
  red[threadIdx.x] = su;
  __syncthreads();
  for (int off = 128; off > 0; off >>= 1) {
    if (threadIdx.x < off) red[threadIdx.x] += red[threadIdx.x + off];
    __syncthreads();
  }
  if (threadIdx.x == 0) { ws[1120 + n] = mx; ws[1136 + n] = red[0]; }
}

// ---------------- pass 5: hidden -> BN/ReLU -> attn -> decode (WMMA) -> fold scatter
__global__ __launch_bounds__(256, 1)
void k5_decode(const float* __restrict__ x, const float* __restrict__ proj_w,
               const float* __restrict__ dec_w, const float* __restrict__ dec_b,
               const float* __restrict__ ws, float* __restrict__ out) {
  __shared__ v2f   sAfrag[4 * 7 * 32];
  __shared__ v2f   sBdec[3 * 16 * 32];   // decode B fragments per (dt, j, lane)
  __shared__ float s_wa[28], s_wb[28];
  __shared__ int   s_geom[28];
  __shared__ float sscale[HID], sbias[HID], sdecb[48];
  int lane = threadIdx.x & 31, wave = threadIdx.x >> 5;
  int hi = lane >> 4;
  int n = blockIdx.x >> 6, chunk = blockIdx.x & 63;
  stage_common(proj_w, ws, n, sAfrag, s_wa, s_wb, s_geom);
  // B fragments: lane ln = col d = dt*16 + (ln&15), k-pair k0 = 4j + 2*(ln>>4)
  for (int idx = threadIdx.x; idx < 3 * 16 * 32; idx += blockDim.x) {
    int dt = idx / 512, rem = idx % 512, j = rem / 32, ln = rem % 32;
    int d  = dt * 16 + (ln & 15);
    int k0 = 4 * j + 2 * (ln >> 4);
    v2f v;
    v.x = (d < DD) ? dec_w[d * HID + k0]     : 0.0f;
    v.y = (d < DD) ? dec_w[d * HID + k0 + 1] : 0.0f;
    sBdec[idx] = v;
  }
  for (int i = threadIdx.x; i < HID; i += blockDim.x) {
    sscale[i] = ws[992 + i];
    sbias[i]  = ws[1056 + i];
  }
  for (int i = threadIdx.x; i < 48; i += blockDim.x)
    sdecb[i] = (i < DD) ? dec_b[i] : 0.0f;
  __syncthreads();
  const float* scores = ws + 1152;
  float mx = ws[1120 + n];
  float rs = 1.0f / ws[1136 + n];
  for (int it = 0; it < 8; ++it) {
    int lbase = ((chunk * 8 + wave) * 8 + it) * 16;
    v8f acc[4];
    hidden_tile(x, sAfrag, s_wa, s_wb, s_geom, n, lbase, lane, hi, acc);
    // BN + ReLU + attention weight (lane's hidden column is lbase + (lane&15))
    float aw = __expf(scores[n * LL + lbase + (lane & 15)] - mx) * rs;
    float yv[4][8];
#pragma unroll
    for (int t = 0; t < 4; ++t)
#pragma unroll
      for (int r = 0; r < 8; ++r) {
        int h = t * 16 + 8 * hi + r;
        float v = fmaf(acc[t][r], sscale[h], sbias[h]);
        yv[t][r] = fmaxf(v, 0.0f) * aw;
      }
    // decode: D[l,d] = sum_h ywT[l,h] * dec_wT[h,d]  (A built from D-layout tile)
    v8f z = 0.0f;
    v8f dacc[3]; dacc[0] = z; dacc[1] = z; dacc[2] = z;
#pragma unroll
    for (int j = 0; j < 16; ++j) {
      const int t  = j >> 2;
      const int s4 = (4 * j) & 15;
      float a0, a1;
      if (s4 < 8) {   // k-chunk lives in low-half lanes
        float o2 = __shfl_xor(yv[t][s4 + 2], 16, 32);
        float o3 = __shfl_xor(yv[t][s4 + 3], 16, 32);
        a0 = hi ? o2 : yv[t][s4];
        a1 = hi ? o3 : yv[t][s4 + 1];
      } else {        // k-chunk lives in high-half lanes
        const int e = s4 - 8;
        float o0 = __shfl_xor(yv[t][e],     16, 32);
        float o1 = __shfl_xor(yv[t][e + 1], 16, 32);
        a0 = hi ? yv[t][e + 2] : o0;
        a1 = hi ? yv[t][e + 3] : o1;
      }
      v2f av = {a0, a1};
#pragma unroll
      for (int dt = 0; dt < 3; ++dt) {
        v2f bv = sBdec[(dt * 16 + j) * 32 + lane];   // ds_load_b64
        dacc[dt] = __builtin_amdgcn_wmma_f32_16x16x4_f32(
            false, av, false, bv, (short)0, dacc[dt], false, false);
      }
    }
    // fold scatter: D layout -> lane col d = dt*16+(lane&15), row l = lbase+8*hi+r
#pragma unroll
    for (int dt = 0; dt < 3; ++dt) {
      int d = dt * 16 + (lane & 15);
      if (d < DD) {
        int co = d / 9, kk = d % 9, ki = kk / 3, kj = kk % 3;
        float db = sdecb[d];
#pragma unroll
        for (int r = 0; r < 8; ++r) {
          int l = lbase + 8 * hi + r;
          int ho = l >> 8, wo = l & 255;
          int hh = ki + 2 * ho - 1, ww = kj + 2 * wo - 1;
          if (hh >= 0 && hh < HH && ww >= 0 && ww < WWI)
            atomicAdd(&out[(((n * COUT + co) * HH) + hh) * WWI + ww],
                      dacc[dt][r] + db);
        }
      }
    }
  }
}

// ---------------- pass 6: tanh(out / (count + eps)), count analytic ----------------
__global__ void k6_final(float* __restrict__ out, int total) {
  for (int idx = blockIdx.x * blockDim.x + threadIdx.x; idx < total;
       idx += gridDim.x * blockDim.x) {
    int w = idx & 511;
    int h = (idx >> 9) & 511;
    float cw = ((w & 1) == 0) ? 1.0f : ((w == 511) ? 1.0f : 2.0f);
    float ch = ((h & 1) == 0) ? 1.0f : ((h == 511) ? 1.0f : 2.0f);
    out[idx] = tanhf(out[idx] / (ch * cw + 1e-6f));
  }
}

extern "C" void kernel_launch(void* const* d_in, const int* in_sizes, int n_in,
                              void* d_out, int out_size, void* d_ws, size_t ws_size,
                              hipStream_t stream) {
  (void)in_sizes; (void)n_in; (void)ws_size;
  const float* x        = (const float*)d_in[0];
  const float* inst_w   = (const float*)d_in[1];
  const float* inst_b   = (const float*)d_in[2];
  const float* proj_w   = (const float*)d_in[3];
  const float* bn_w     = (const float*)d_in[4];
  const float* bn_b     = (const float*)d_in[5];
  const float* scorer_w = (const float*)d_in[6];
  const float* scorer_b = (const float*)d_in[7];
  const float* dec_w    = (const float*)d_in[8];
  const float* dec_b    = (const float*)d_in[9];
  float* out = (float*)d_out;
  float* ws  = (float*)d_ws;

  k0_zero   <<<4096, 256, 0, stream>>>(out, out_size, ws);
  k1_inst   <<<NB * PD, 256, 0, stream>>>(x, inst_w, inst_b, ws);
  k2_bnstats<<<1024, 256, 0, stream>>>(x, proj_w, ws);
  k2b_bnfin <<<1, 64, 0, stream>>>(bn_w, bn_b, ws);
  k3_scores <<<1024, 256, 0, stream>>>(x, proj_w, scorer_w, scorer_b, ws);
  k4_softmax<<<NB, 256, 0, stream>>>(ws);
  k5_decode <<<1024, 256, 0, stream>>>(x, proj_w, dec_w, dec_b, ws, out);
  k6_final  <<<16384, 256, 0, stream>>>(out, out_size);
}